// MyLoss_19619410608500
// MI455X (gfx1250) — compile-verified
//
#include <hip/hip_runtime.h>
#include <hip/hip_bf16.h>

#define NVOX 8000
#define NPAD 8192
#define EPSV 0.1f
#define INVE 10.0f
#define NEGV -1.0e9f
#define NITER 30

typedef __attribute__((ext_vector_type(2))) float v2f;
typedef __attribute__((ext_vector_type(8))) float v8f;

// ---------- reduction helpers (wave32, deterministic order) ----------
__device__ __forceinline__ float wave_sum32(float v) {
#pragma unroll
  for (int o = 16; o > 0; o >>= 1) v += __shfl_xor(v, o, 32);
  return v;
}
__device__ __forceinline__ float wave_max32(float v) {
#pragma unroll
  for (int o = 16; o > 0; o >>= 1) v = fmaxf(v, __shfl_xor(v, o, 32));
  return v;
}
// full-block (256 thr) sum; result valid on ALL threads
__device__ __forceinline__ float block_sum256(float v, float* red) {
  v = wave_sum32(v);
  __syncthreads();
  if ((threadIdx.x & 31) == 0) red[threadIdx.x >> 5] = v;
  __syncthreads();
  float s = 0.f;
#pragma unroll
  for (int w = 0; w < 8; ++w) s += red[w];
  return s;
}

// ---------- tiny setup / finish kernels ----------
__global__ __launch_bounds__(256) void partial_weights(const float* __restrict__ p,
                                                       const float* __restrict__ t0,
                                                       float* __restrict__ part) {
  __shared__ float red[8];
  float s1 = 0.f, s2 = 0.f;
  for (int j = blockIdx.x * 256 + threadIdx.x; j < NVOX; j += 32 * 256) {
    float tv = t0[j], pv = p[j];
    s1 += (tv != 0.0f) ? tv : 0.0f;
    s2 += (pv > 100.0f) ? pv : 0.0f;
  }
  float r1 = block_sum256(s1, red);
  __syncthreads();
  float r2 = block_sum256(s2, red);
  if (threadIdx.x == 0) { part[blockIdx.x] = r1; part[32 + blockIdx.x] = r2; }
}

__global__ void sum32(const float* __restrict__ part, float* __restrict__ dst) {
  float v = part[threadIdx.x];
  v = wave_sum32(v);
  if (threadIdx.x == 0) dst[0] = v;
}

__global__ __launch_bounds__(256) void make_logw(const float* __restrict__ p,
                                                 const float* __restrict__ t0,
                                                 const float* __restrict__ scal,
                                                 float* __restrict__ la, float* __restrict__ aw,
                                                 float* __restrict__ lb, float* __restrict__ bw) {
  int j = blockIdx.x * 256 + threadIdx.x;
  if (j >= NVOX) return;
  float S1 = fmaxf(scal[0], 1e-30f), S2 = fmaxf(scal[1], 1e-30f);
  float tv = t0[j], pv = p[j];
  float w1 = (tv != 0.0f) ? tv : 0.0f;
  float w2 = (pv > 100.0f) ? pv : 0.0f;
  float an = w1 / S1, bn = w2 / S2;
  aw[j] = an; bw[j] = bn;
  la[j] = (an > 0.0f) ? __logf(fmaxf(an, 1e-30f)) : NEGV;
  lb[j] = (bn > 0.0f) ? __logf(fmaxf(bn, 1e-30f)) : NEGV;
}

__global__ __launch_bounds__(256) void zero_fg(float* __restrict__ f, float* __restrict__ g) {
  int j = blockIdx.x * 256 + threadIdx.x;
  if (j < NVOX) { f[j] = 0.0f; g[j] = 0.0f; }
}

// ---------- the hot kernel: one Sinkhorn half-update (one row per block) ----------
// dual_out[i] = -EPS * LSE_j( lw[j] + (dual_in[j] - C(i,j)) / EPS )
// C(i,j) recomputed from the 20^3 integer lattice (no 256MB matrix traffic;
// lw/dual_in are 32KB each and stay hot in L2 across all 8000 blocks).
__global__ __launch_bounds__(256) void sinkhorn_row(const float* __restrict__ lw,
                                                    const float* __restrict__ dual_in,
                                                    float* __restrict__ dual_out) {
  __shared__ float v[NPAD];
  __shared__ float red[16];
  const int i = blockIdx.x;
  const int tid = threadIdx.x;
  const int wave = tid >> 5, lane = tid & 31;

  const int xi = i / 400;
  const int ri = i - 400 * xi;
  const int yi = ri / 20;
  const int zi = ri - 20 * yi;

  // ---- pass 1: fill v[] in LDS, track per-thread max ----
  float m = -3.0e38f;
  for (int j = tid; j < NPAD; j += 256) {
    float val;
    if (j < NVOX) {
      int xj = j / 400;
      int rj = j - 400 * xj;
      int yj = rj / 20;
      int zj = rj - 20 * yj;
      int dx = xi - xj, dy = yi - yj, dz = zi - zj;
      int d2 = dx * dx + dy * dy + dz * dz;
      float c = __builtin_amdgcn_sqrtf((float)d2);   // v_sqrt_f32
      val = lw[j] + INVE * dual_in[j] - INVE * c;
    } else {
      val = -3.0e38f;   // padding: exp -> 0, never the max
    }
    v[j] = val;
    m = fmaxf(m, val);
  }
  m = wave_max32(m);
  if (lane == 0) red[wave] = m;
  __syncthreads();
  float mb = red[0];
#pragma unroll
  for (int w = 1; w < 8; ++w) mb = fmaxf(mb, red[w]);
  __syncthreads();

  // ---- pass 2: sum exp(v - mb) via V_WMMA_F32_16X16X4_F32 with all-ones B ----
  // Each WMMA folds a 16x4 tile (64 exp values) into an f32 accumulator; two
  // interleaved accumulators break the WMMA->WMMA D->C RAW chain so the
  // scheduler can fill the latency with v_exp_f32 work instead of v_nops.
  // Every D column equals the row sums, so wave total = 32-lane reduce / 16.
  v8f acc0 = {};
  v8f acc1 = {};
  v2f ones; ones.x = 1.0f; ones.y = 1.0f;
  const int base = wave * (NPAD / 8);            // 1024 values per wave
#pragma unroll
  for (int step = 0; step < (NPAD / 8) / 128; ++step) {  // 8 x 2 WMMA per wave
    int idx = base + step * 128 + 2 * lane;
    v2f a0, a1;
    a0.x = __expf(v[idx] - mb);                   // v_exp_f32 on trans pipe
    a0.y = __expf(v[idx + 1] - mb);
    a1.x = __expf(v[idx + 64] - mb);
    a1.y = __expf(v[idx + 65] - mb);
    acc0 = __builtin_amdgcn_wmma_f32_16x16x4_f32(false, a0, false, ones,
                                                 (short)0, acc0, false, false);
    acc1 = __builtin_amdgcn_wmma_f32_16x16x4_f32(false, a1, false, ones,
                                                 (short)0, acc1, false, false);
  }
  v8f acc;
#pragma unroll
  for (int e = 0; e < 8; ++e) acc[e] = acc0[e] + acc1[e];
  float s = acc[0] + acc[1] + acc[2] + acc[3] + acc[4] + acc[5] + acc[6] + acc[7];
  s = wave_sum32(s) * (1.0f / 16.0f);
  if (lane == 0) red[wave] = s;
  __syncthreads();
  if (tid == 0) {
    float st = red[0] + red[1] + red[2] + red[3] + red[4] + red[5] + red[6] + red[7];
    dual_out[i] = -EPSV * (mb + __logf(st));
  }
}

// ---------- OT value <a,f> + <b,g> ----------
__global__ __launch_bounds__(256) void ot_partial(const float* __restrict__ a,
                                                  const float* __restrict__ f,
                                                  const float* __restrict__ b,
                                                  const float* __restrict__ g,
                                                  float* __restrict__ part) {
  __shared__ float red[8];
  float s = 0.f;
  for (int j = blockIdx.x * 256 + threadIdx.x; j < NVOX; j += 32 * 256)
    s += a[j] * f[j] + b[j] * g[j];
  float r = block_sum256(s, red);
  if (threadIdx.x == 0) part[blockIdx.x] = r;
}

// ---------- pointwise CE + masked MSE partials (8 channels) ----------
__global__ __launch_bounds__(256) void pointwise_partial(const float* __restrict__ p,
                                                         const float* __restrict__ t0,
                                                         float* __restrict__ part) {
  __shared__ float red[8];
  float ch[8];
#pragma unroll
  for (int c = 0; c < 8; ++c) ch[c] = 0.f;
  for (int j = blockIdx.x * 256 + threadIdx.x; j < NVOX; j += 32 * 256) {
    float pv = p[j], tv = t0[j];
    float pc = fminf(fmaxf(pv, 0.0f), 1.0f);
    float l0 = 1.0f - pc, l1 = pc;
    float mm = fmaxf(l0, l1);
    float lse = mm + __logf(__expf(l0 - mm) + __expf(l1 - mm));
    int tgt = (int)fminf(fmaxf(tv, 0.0f), 1.0f);   // truncation like .long()
    float nll = lse - (tgt ? l1 : l0);
    float cw = tgt ? 1.0f : 0.001f;
    ch[0] += cw * nll;
    ch[1] += cw;
    float d = pv - tv, d2 = d * d;
    if (tv > 0.0f)      { ch[2] += d2; ch[3] += 1.0f; }
    else                { ch[4] += d2; ch[5] += 1.0f; }
    if (tv > 2000.0f)   { ch[6] += d2; ch[7] += 1.0f; }
  }
#pragma unroll
  for (int c = 0; c < 8; ++c) {
    float r = block_sum256(ch[c], red);
    if (threadIdx.x == 0) part[c * 32 + blockIdx.x] = r;
    __syncthreads();
  }
}

__global__ void final_combine(const float* __restrict__ part,
                              const float* __restrict__ scal,
                              float* __restrict__ out) {
  __shared__ float ch[8];
  int tid = threadIdx.x;
  if (tid < 8) {
    float s = 0.f;
    for (int b = 0; b < 32; ++b) s += part[tid * 32 + b];
    ch[tid] = s;
  }
  __syncthreads();
  if (tid == 0) {
    float wass = fabsf(scal[2] - 0.5f * scal[3] - 0.5f * scal[4]) * 0.625f;
    float ce   = (ch[0] / ch[1]) * 1.0e8f * 1.1f;
    float spur = ch[2] / fmaxf(ch[3], 1.0f) * 10000.0f;
    float lb_  = ch[4] / fmaxf(ch[5], 1.0f) * 25000.0f;
    float lmx  = ch[6] / fmaxf(ch[7], 1.0f) * 1000.0f;
    out[0] = wass + ce + lb_ + spur + lmx;
  }
}

// ---------- host orchestration ----------
extern "C" void kernel_launch(void* const* d_in, const int* in_sizes, int n_in,
                              void* d_out, int out_size, void* d_ws, size_t ws_size,
                              hipStream_t stream) {
  const float* p  = (const float*)d_in[0];
  const float* t0 = (const float*)d_in[1];   // t is [1, 8000] -> flat 8000
  float* ws = (float*)d_ws;

  float* LA   = ws;             // 8000
  float* LB   = ws + 8000;      // 8000
  float* Aw   = ws + 16000;     // 8000
  float* Bw   = ws + 24000;     // 8000
  float* F    = ws + 32000;     // 8000
  float* G    = ws + 40000;     // 8000
  float* PART = ws + 48000;     // 256
  float* SCAL = ws + 48256;     // 0:S1 1:S2 2:ot_ab 3:ot_aa 4:ot_bb
  float* out  = (float*)d_out;

  partial_weights<<<32, 256, 0, stream>>>(p, t0, PART);
  sum32<<<1, 32, 0, stream>>>(PART, SCAL + 0);
  sum32<<<1, 32, 0, stream>>>(PART + 32, SCAL + 1);
  make_logw<<<32, 256, 0, stream>>>(p, t0, SCAL, LA, Aw, LB, Bw);

  const float* pLA[3] = { LA, LA, LB };   // row-side log weights (a)
  const float* pLB[3] = { LB, LA, LB };   // col-side log weights (b)
  const float* pA[3]  = { Aw, Aw, Bw };
  const float* pB[3]  = { Bw, Aw, Bw };

  for (int k = 0; k < 3; ++k) {
    zero_fg<<<32, 256, 0, stream>>>(F, G);
    for (int it = 0; it < NITER; ++it) {
      sinkhorn_row<<<NVOX, 256, 0, stream>>>(pLB[k], G, F);   // f-update
      sinkhorn_row<<<NVOX, 256, 0, stream>>>(pLA[k], F, G);   // g-update
    }
    ot_partial<<<32, 256, 0, stream>>>(pA[k], F, pB[k], G, PART);
    sum32<<<1, 32, 0, stream>>>(PART, SCAL + 2 + k);
  }

  pointwise_partial<<<32, 256, 0, stream>>>(p, t0, PART);
  final_combine<<<1, 64, 0, stream>>>(PART, SCAL, out);
}